// Phi_81595788689756
// MI455X (gfx1250) — compile-verified
//
#include <hip/hip_runtime.h>
#include <math.h>

typedef float v2f __attribute__((ext_vector_type(2)));
typedef float v8f __attribute__((ext_vector_type(8)));
typedef int   v4i __attribute__((ext_vector_type(4)));

// Fast, accurate-enough activation: exp of non-positive arg, log of (1,2].
// Hardware v_exp_f32 / v_log_f32 are ~1 ulp -> absolute error ~1e-7 on the sum.
__device__ __forceinline__ float antideriv_tanh_dev(float z) {
    float ax = fabsf(z);
    return ax + __logf(1.0f + __expf(-2.0f * ax));
}

// ---- CDNA5 async global->LDS staging (ASYNCcnt-tracked), with fallback ----
#if __has_builtin(__builtin_amdgcn_global_load_async_to_lds_b128)
#define HAVE_ASYNC_LDS 1
__device__ __forceinline__ void async_cp16(const float* g, float* l) {
    __builtin_amdgcn_global_load_async_to_lds_b128(
        (__attribute__((address_space(1))) v4i*)(g),
        (__attribute__((address_space(3))) v4i*)(l), 0, 0);
}
__device__ __forceinline__ void wait_async0() {
#if __has_builtin(__builtin_amdgcn_s_wait_asynccnt)
    __builtin_amdgcn_s_wait_asynccnt(0);
#else
    asm volatile("s_wait_asynccnt 0x0" ::: "memory");
#endif
}
#else
#define HAVE_ASYNC_LDS 0
#endif

// ---------------------------------------------------------------------------
// Fused NT-GEMM:  Out = [Ures + h *] antideriv_tanh( Ag @ Wg^T + bias )
//   Ag : M x K row-major (activations);  Wg : N x K row-major (weights)
//   V_WMMA_F32_16X16X4_F32 = exact fp32 tensor path (reference is fp32).
// Block 128x128, BK=16, 256 threads = 8 waves (2x4); 64x32 tile/wave = 8 v8f.
// ---------------------------------------------------------------------------
#define BM 128
#define BN 128
#define BK 16
#define LDT 20   // padded LDS row stride (floats): 16B-aligned rows, no bank conflicts

__global__ __launch_bounds__(256, 2) void gemm_res_act_kernel(
    const float* __restrict__ Ag, const float* __restrict__ Wg,
    const float* __restrict__ bias, const float* __restrict__ Ures,
    float* __restrict__ Out, int M, int N, int K, float hstep, int residual)
{
    __shared__ float sA[2][BM * LDT];
    __shared__ float sB[2][BN * LDT];

    const int tid  = threadIdx.x;
    const int lane = tid & 31;
    const int wave = tid >> 5;
    const int wm   = wave >> 2;   // 0..1 : wave row in block
    const int wn   = wave & 3;    // 0..3 : wave col in block
    const int bm   = blockIdx.x * BM;
    const int bn   = blockIdx.y * BN;

    // staging decomposition: thread t handles rows (t>>2) and (t>>2)+64, 16B each
    const int lr = tid >> 2;            // 0..63
    const int lc = (tid & 3) * 4;       // 0,4,8,12 (floats)
    const float* aPtr = Ag + (size_t)(bm + lr) * K + lc;
    const float* bPtr = Wg + (size_t)(bn + lr) * K + lc;
    const size_t rowHop = (size_t)64 * K;

    // WMMA fragment lane decomposition (ISA 7.12.2: 32-bit A 16x4 / B 4x16)
    const int lm = lane & 15;           // row (M for A, N for B) within 16
    const int lk = (lane >> 4) << 1;    // k offset: 0 or 2

    v8f acc[4][2] = {};
    const int nK = K / BK;

#if HAVE_ASYNC_LDS
    // ---- async-staged double-buffered pipeline ----
    {   // stage tile 0 -> buf 0
        async_cp16(aPtr,          &sA[0][lr * LDT + lc]);
        async_cp16(aPtr + rowHop, &sA[0][(lr + 64) * LDT + lc]);
        async_cp16(bPtr,          &sB[0][lr * LDT + lc]);
        async_cp16(bPtr + rowHop, &sB[0][(lr + 64) * LDT + lc]);
        wait_async0();
        __syncthreads();
    }
#else
    float4 ra0 = *(const float4*)(aPtr);
    float4 ra1 = *(const float4*)(aPtr + rowHop);
    float4 rb0 = *(const float4*)(bPtr);
    float4 rb1 = *(const float4*)(bPtr + rowHop);
    *(float4*)&sA[0][lr * LDT + lc]        = ra0;
    *(float4*)&sA[0][(lr + 64) * LDT + lc] = ra1;
    *(float4*)&sB[0][lr * LDT + lc]        = rb0;
    *(float4*)&sB[0][(lr + 64) * LDT + lc] = rb1;
    __syncthreads();
#endif

    for (int kt = 0; kt < nK; ++kt) {
        const int cur = kt & 1;
#if HAVE_ASYNC_LDS
        if (kt + 1 < nK) {  // kick off async fill of the other buffer
            const int nxt = cur ^ 1;
            const float* ap = aPtr + (size_t)(kt + 1) * BK;
            const float* bp = bPtr + (size_t)(kt + 1) * BK;
            async_cp16(ap,          &sA[nxt][lr * LDT + lc]);
            async_cp16(ap + rowHop, &sA[nxt][(lr + 64) * LDT + lc]);
            async_cp16(bp,          &sB[nxt][lr * LDT + lc]);
            async_cp16(bp + rowHop, &sB[nxt][(lr + 64) * LDT + lc]);
        }
#else
        if (kt + 1 < nK) {
            const float* ap = aPtr + (size_t)(kt + 1) * BK;
            const float* bp = bPtr + (size_t)(kt + 1) * BK;
            ra0 = *(const float4*)(ap);
            ra1 = *(const float4*)(ap + rowHop);
            rb0 = *(const float4*)(bp);
            rb1 = *(const float4*)(bp + rowHop);
        }
#endif
        #pragma unroll
        for (int kk = 0; kk < BK; kk += 4) {
            v2f afr[4], bfr[2];
            #pragma unroll
            for (int s = 0; s < 4; ++s) {
                const int row = wm * 64 + s * 16 + lm;
                afr[s] = *(const v2f*)&sA[cur][row * LDT + kk + lk];
            }
            #pragma unroll
            for (int t = 0; t < 2; ++t) {
                const int col = wn * 32 + t * 16 + lm;
                bfr[t] = *(const v2f*)&sB[cur][col * LDT + kk + lk];
            }
            #pragma unroll
            for (int s = 0; s < 4; ++s)
                #pragma unroll
                for (int t = 0; t < 2; ++t)
                    acc[s][t] = __builtin_amdgcn_wmma_f32_16x16x4_f32(
                        false, afr[s], false, bfr[t], (short)0, acc[s][t],
                        false, false);
        }
        if (kt + 1 < nK) {
#if HAVE_ASYNC_LDS
            wait_async0();          // our async writes into nxt have landed
            __syncthreads();        // everyone's writes visible to everyone
#else
            const int nxt = cur ^ 1;
            *(float4*)&sA[nxt][lr * LDT + lc]        = ra0;
            *(float4*)&sA[nxt][(lr + 64) * LDT + lc] = ra1;
            *(float4*)&sB[nxt][lr * LDT + lc]        = rb0;
            *(float4*)&sB[nxt][(lr + 64) * LDT + lc] = rb1;
            __syncthreads();
#endif
        }
    }

    // Epilogue: bias + activation (+ residual).  C/D layout:
    // VGPR r, lanes 0-15 -> M=r, N=lane ; lanes 16-31 -> M=r+8, N=lane-16
    const int rofs = (lane >> 4) * 8;
    #pragma unroll
    for (int s = 0; s < 4; ++s) {
        #pragma unroll
        for (int t = 0; t < 2; ++t) {
            const int col = bn + wn * 32 + t * 16 + (lane & 15);
            const float bval = bias[col];
            #pragma unroll
            for (int r = 0; r < 8; ++r) {
                const int row = bm + wm * 64 + s * 16 + rofs + r;
                const float g = antideriv_tanh_dev(acc[s][t][r] + bval);
                const size_t idx = (size_t)row * N + col;
                Out[idx] = residual ? (Ures[idx] + hstep * g) : g;
            }
        }
    }
}

// ---------------------------------------------------------------------------
// quad_kernel: ws[i] = 0.5*||A x_i||^2 + dot(x_i, cw) + cb   (x^T A^T A x = ||Ax||^2)
// ---------------------------------------------------------------------------
__global__ __launch_bounds__(256) void quad_kernel(
    const float* __restrict__ x, const float* __restrict__ A,
    const float* __restrict__ cw, const float* __restrict__ cb,
    float* __restrict__ quadOut, int K)
{
    const int lane = threadIdx.x & 31;
    const int wave = threadIdx.x >> 5;
    const int row  = blockIdx.x * 8 + wave;
    const float* xr = x + (size_t)row * K;

    float accA[10] = {0.f,0.f,0.f,0.f,0.f,0.f,0.f,0.f,0.f,0.f};
    float accc = 0.f;
    for (int k = lane; k < K; k += 32) {
        const float xv = xr[k];
        accc += xv * cw[k];
        #pragma unroll
        for (int r = 0; r < 10; ++r) accA[r] += xv * A[(size_t)r * K + k];
    }
    #pragma unroll
    for (int off = 16; off > 0; off >>= 1) {
        accc += __shfl_xor(accc, off, 32);
        #pragma unroll
        for (int r = 0; r < 10; ++r) accA[r] += __shfl_xor(accA[r], off, 32);
    }
    if (lane == 0) {
        float q = 0.f;
        #pragma unroll
        for (int r = 0; r < 10; ++r) q += accA[r] * accA[r];
        quadOut[row] = 0.5f * q + accc + cb[0];
    }
}

// ---------------------------------------------------------------------------
// final_kernel: out[i] = quad[i] + dot(u_i, ww)  (single write to d_out -> replay-safe)
// ---------------------------------------------------------------------------
__global__ __launch_bounds__(256) void final_kernel(
    const float* __restrict__ U, const float* __restrict__ ww,
    const float* __restrict__ quadIn, float* __restrict__ out, int K)
{
    const int lane = threadIdx.x & 31;
    const int wave = threadIdx.x >> 5;
    const int row  = blockIdx.x * 8 + wave;
    const float* ur = U + (size_t)row * K;
    float acc = 0.f;
    for (int k = lane; k < K; k += 32) acc += ur[k] * ww[k];
    #pragma unroll
    for (int off = 16; off > 0; off >>= 1) acc += __shfl_xor(acc, off, 32);
    if (lane == 0) out[row] = quadIn[row] + acc;
}

extern "C" void kernel_launch(void* const* d_in, const int* in_sizes, int n_in,
                              void* d_out, int out_size, void* d_ws, size_t ws_size,
                              hipStream_t stream)
{
    (void)n_in; (void)out_size; (void)ws_size;
    const float* x  = (const float*)d_in[0];
    const float* A  = (const float*)d_in[1];
    const float* W0 = (const float*)d_in[2];
    const float* b0 = (const float*)d_in[3];
    const float* Wh = (const float*)d_in[4];
    const float* bh = (const float*)d_in[5];
    const float* ww = (const float*)d_in[6];
    const float* cw = (const float*)d_in[7];
    const float* cb = (const float*)d_in[8];
    float* out = (float*)d_out;

    const int m   = in_sizes[3];              // 2048
    const int dp1 = in_sizes[7];              // 2048
    const int M   = in_sizes[0] / dp1;        // 16384
    const int NB  = in_sizes[5] / m;          // 3 residual blocks
    const float h = 1.0f / (float)NB;

    const size_t uBytes = (size_t)M * m * sizeof(float);
    float* U0      = (float*)d_ws;
    float* U1      = (float*)((char*)d_ws + uBytes);
    float* quadBuf = (float*)((char*)d_ws + 2 * uBytes);

    quad_kernel<<<M / 8, 256, 0, stream>>>(x, A, cw, cb, quadBuf, dp1);

    dim3 grid(M / BM, m / BN);
    gemm_res_act_kernel<<<grid, 256, 0, stream>>>(x, W0, b0, nullptr, U0,
                                                  M, m, dp1, 0.f, 0);
    const float* uin = U0;
    float* uout = U1;
    for (int i = 0; i < NB; ++i) {
        gemm_res_act_kernel<<<grid, 256, 0, stream>>>(
            uin, Wh + (size_t)i * m * m, bh + (size_t)i * m,
            uin, uout, M, m, m, h, 1);
        const float* tmp = uin; uin = uout; uout = (float*)tmp;
    }
    final_kernel<<<M / 8, 256, 0, stream>>>(uin, ww, quadBuf, out, m);
}